// SpectralConv2d_58755152609653
// MI455X (gfx1250) — compile-verified
//
#include <hip/hip_runtime.h>
#include <math.h>

// SpectralConv2d (pole-residue FNO) for gfx1250.
// All O(N^3)-shaped contractions run on V_WMMA_F32_16X16X4_F32:
//   - fft2 / ifft2 as 128x128 DFT-matrix GEMMs (complex = 4 real WMMAs,
//     real-output = 2; F32 WMMA has no A/B NEG so imag operands are negated
//     in VALU),
//   - T1[b,i,(k,p),x] = sum_o A1*alpha  (complex GEMM per (b,i) plane),
//   - x2[b,k,y,x']    = Re(sum_{c,p} E1*G2)  (real-out GEMM per (b,k) plane).
// Small-inner-dim pole contractions stay on VALU. Everything fp32; working
// set (~160 MB ws) fits in the 192 MB L2 so the pipeline is compute-shaped.

typedef __attribute__((ext_vector_type(2))) float v2f;
typedef __attribute__((ext_vector_type(8))) float v8f;

#define NB   8
#define NC   32
#define NN   128
#define NM1  4
#define NM2  5
#define NP   (NN * NN)     // 16384
#define NIMG (NB * NC)     // 256
#define TWO_PI_F 6.283185307179586f

__device__ __forceinline__ float2 cfma(float2 a, float2 b, float2 c) {
  c.x = fmaf(a.x, b.x, fmaf(-a.y, b.y, c.x));
  c.y = fmaf(a.x, b.y, fmaf(a.y, b.x, c.y));
  return c;
}

__device__ __forceinline__ v8f wmma4(v2f a, v2f b, v8f c) {
  // D = A(16x4, f32) * B(4x16, f32) + C(16x16, f32), wave32, EXEC all ones.
  return __builtin_amdgcn_wmma_f32_16x16x4_f32(false, a, false, b, (short)0, c,
                                               false, false);
}

// ---------------------------------------------------------------------------
// Tables: DFT twiddles, pole factors A1/A2, spatial exponentials E1/E2.
// Wf[k,n] = e^{-2pi i kn/N};  Wi[k,n] = e^{+2pi i kn/N}/N
// A1[i,k,p,o] = 1/(i*2pi*f1[o] - wp1[i,k,p])
// E1[i,k,p,y] = exp(wp1*ty[y]) / N^2   (x2 scale folded in)
// E2[i,k,q,x] = exp(wp2*tx[x])
// ---------------------------------------------------------------------------
__global__ __launch_bounds__(256) void k_tables(
    const float* __restrict__ wp1_re, const float* __restrict__ wp1_im,
    const float* __restrict__ wp2_re, const float* __restrict__ wp2_im,
    const float* __restrict__ ty, const float* __restrict__ tx,
    float* __restrict__ Wf_re, float* __restrict__ Wf_im,
    float* __restrict__ Wi_re, float* __restrict__ Wi_im,
    float2* __restrict__ A1, float2* __restrict__ A2,
    float2* __restrict__ E1, float2* __restrict__ E2) {
  int idx = blockIdx.x * 256 + threadIdx.x;

  if (idx < NP) {
    int k = idx >> 7, n = idx & 127;
    int m = (k * n) & 127;  // exact modular phase
    float ang = TWO_PI_F * (float)m * (1.0f / 128.0f);
    float c = cosf(ang), s = sinf(ang);
    Wf_re[idx] = c;
    Wf_im[idx] = -s;
    Wi_re[idx] = c * (1.0f / 128.0f);
    Wi_im[idx] = s * (1.0f / 128.0f);
  }

  if (idx < NC * NC * NM1 * NN) {  // 524288: A1 and E1 share index space
    int o = idx & 127;
    int p = (idx >> 7) & 3;
    int ik = idx >> 9;  // i*32+k
    float wr = wp1_re[ik * NM1 + p], wi = wp1_im[ik * NM1 + p];
    float dty = ty[1] - ty[0];
    int kk = (o < 64) ? o : o - 128;  // fftfreq numerator
    float lam = TWO_PI_F * (float)kk / (128.0f * dty);
    float a = -wr, b = lam - wi;  // lam1 - wp1 = (-wr) + i(lam - wi)
    float inv = 1.0f / (a * a + b * b);
    A1[idx] = make_float2(a * inv, -b * inv);
    float t = ty[o];
    float mag = expf(wr * t) * (1.0f / (float)NP);
    E1[idx] = make_float2(mag * cosf(wi * t), mag * sinf(wi * t));
  }

  if (idx < NC * NC * NM2 * NN) {  // 655360: A2 and E2
    int x = idx & 127;
    int q = (idx >> 7) % NM2;
    int ik = idx / (NM2 * NN);
    float wr = wp2_re[ik * NM2 + q], wi = wp2_im[ik * NM2 + q];
    float dtx = tx[1] - tx[0];
    int kk = (x < 64) ? x : x - 128;
    float lam = TWO_PI_F * (float)kk / (128.0f * dtx);
    float a = -wr, b = lam - wi;
    float inv = 1.0f / (a * a + b * b);
    A2[idx] = make_float2(a * inv, -b * inv);
    float t = tx[x];
    float mag = expf(wr * t);
    E2[idx] = make_float2(mag * cosf(wi * t), mag * sinf(wi * t));
  }
}

// W2[i,k,p,x] = sum_q wr[i,k,p,q] * A2[i,k,q,x]  (pre-contraction for Hsum)
__global__ __launch_bounds__(256) void k_W2(const float2* __restrict__ A2,
                                            const float* __restrict__ wr_re,
                                            const float* __restrict__ wr_im,
                                            float2* __restrict__ W2) {
  const int idx = blockIdx.x * 256 + threadIdx.x;  // < 524288
  const int x = idx & 127;
  const int p = (idx >> 7) & 3;
  const int ik = idx >> 9;
  float2 acc = make_float2(0.f, 0.f);
  for (int q = 0; q < NM2; ++q) {
    float2 w = make_float2(wr_re[ik * 20 + p * 5 + q],
                           wr_im[ik * 20 + p * 5 + q]);
    acc = cfma(w, A2[(ik * NM2 + q) * NN + x], acc);
  }
  W2[idx] = acc;
}

// ---------------------------------------------------------------------------
// Forward FFT pass A (rows): tmp[y,k2] = sum_n x[y,n] * Wf[k2,n]   (x real)
// ---------------------------------------------------------------------------
__global__ __launch_bounds__(32) void k_fwd_rows(
    const float* __restrict__ X, const float* __restrict__ Wre,
    const float* __restrict__ Wim, float2* __restrict__ out) {
  const int img = blockIdx.y;
  const int r0 = (blockIdx.x >> 3) * 16, c0 = (blockIdx.x & 7) * 16;
  const int l = threadIdx.x;
  const int l16 = l & 15;
  const int koff = (l >> 4) * 2;  // 32-bit A/B layout: lanes16-31 hold K+2,K+3
  const float* Xp = X + (size_t)img * NP;
  v8f cre = {}, cim = {};
  for (int t0 = 0; t0 < NN; t0 += 4) {
    const int t = t0 + koff;
    v2f a;
    a.x = Xp[(r0 + l16) * NN + t];
    a.y = Xp[(r0 + l16) * NN + t + 1];
    v2f bre, bim;
    bre.x = Wre[(c0 + l16) * NN + t];
    bre.y = Wre[(c0 + l16) * NN + t + 1];
    bim.x = Wim[(c0 + l16) * NN + t];
    bim.y = Wim[(c0 + l16) * NN + t + 1];
    cre = wmma4(a, bre, cre);
    cim = wmma4(a, bim, cim);
  }
  float2* outp = out + (size_t)img * NP;
  const int hi = (l >> 4) * 8;  // C layout: VGPR v -> row v (+8 for hi lanes)
  for (int v = 0; v < 8; ++v)
    outp[(r0 + v + hi) * NN + c0 + l16] = make_float2(cre[v], cim[v]);
}

// ---------------------------------------------------------------------------
// Complex GEMM: C[r,c] = sum_t (Are+iAim)[r,t] * B[t,c]   (B interleaved f2)
// Forward pass B (A=Wf) and inverse pass A (A=Wi).
// ---------------------------------------------------------------------------
__global__ __launch_bounds__(32) void k_gemm_cc(
    const float* __restrict__ Are, const float* __restrict__ Aim,
    const float2* __restrict__ B, float2* __restrict__ C) {
  const int img = blockIdx.y;
  const int r0 = (blockIdx.x >> 3) * 16, c0 = (blockIdx.x & 7) * 16;
  const int l = threadIdx.x;
  const int l16 = l & 15;
  const int koff = (l >> 4) * 2;
  const float2* Bp = B + (size_t)img * NP;
  v8f crr = {}, cii = {};
  for (int t0 = 0; t0 < NN; t0 += 4) {
    const int t = t0 + koff;
    v2f are, aim;
    are.x = Are[(r0 + l16) * NN + t];
    are.y = Are[(r0 + l16) * NN + t + 1];
    aim.x = Aim[(r0 + l16) * NN + t];
    aim.y = Aim[(r0 + l16) * NN + t + 1];
    float2 b0 = Bp[t * NN + c0 + l16];
    float2 b1 = Bp[(t + 1) * NN + c0 + l16];
    v2f bre, bim, nbim;
    bre.x = b0.x;   bre.y = b1.x;
    bim.x = b0.y;   bim.y = b1.y;
    nbim.x = -b0.y; nbim.y = -b1.y;  // F32 WMMA has no A/B NEG; negate in VALU
    crr = wmma4(are, bre, crr);
    crr = wmma4(aim, nbim, crr);
    cii = wmma4(are, bim, cii);
    cii = wmma4(aim, bre, cii);
  }
  float2* Cp = C + (size_t)img * NP;
  const int hi = (l >> 4) * 8;
  for (int v = 0; v < 8; ++v)
    Cp[(r0 + v + hi) * NN + c0 + l16] = make_float2(crr[v], cii[v]);
}

// ---------------------------------------------------------------------------
// T1 as WMMA GEMM. For plane (b,i): rows r = k*4+p (128 of them):
//   T1[r][x] = sum_o A1_i[r][o] * alpha_{b,i}[o][x]   (complex x complex)
// A1 layout [i][r][o] is already the row-major A operand.
// ---------------------------------------------------------------------------
__global__ __launch_bounds__(32) void k_T1_gemm(
    const float2* __restrict__ A1, const float2* __restrict__ alpha,
    float2* __restrict__ T1) {
  const int plane = blockIdx.y;  // b*32+i
  const int i = plane & 31;
  const int r0 = (blockIdx.x >> 3) * 16, c0 = (blockIdx.x & 7) * 16;
  const int l = threadIdx.x;
  const int l16 = l & 15;
  const int koff = (l >> 4) * 2;
  const float2* Ap = A1 + (size_t)i * NP;         // [r][o]
  const float2* Bp = alpha + (size_t)plane * NP;  // [o][x]
  v8f crr = {}, cii = {};
  for (int t0 = 0; t0 < NN; t0 += 4) {
    const int t = t0 + koff;
    float2 a0 = Ap[(r0 + l16) * NN + t];
    float2 a1 = Ap[(r0 + l16) * NN + t + 1];
    float2 b0 = Bp[t * NN + c0 + l16];
    float2 b1 = Bp[(t + 1) * NN + c0 + l16];
    v2f are, aim, bre, bim, nbim;
    are.x = a0.x;   are.y = a1.x;
    aim.x = a0.y;   aim.y = a1.y;
    bre.x = b0.x;   bre.y = b1.x;
    bim.x = b0.y;   bim.y = b1.y;
    nbim.x = -b0.y; nbim.y = -b1.y;
    crr = wmma4(are, bre, crr);
    crr = wmma4(aim, nbim, crr);
    cii = wmma4(are, bim, cii);
    cii = wmma4(aim, bre, cii);
  }
  float2* Cp = T1 + (size_t)plane * NP;  // [r][x] == [b,i,k,p,x]
  const int hi = (l >> 4) * 8;
  for (int v = 0; v < 8; ++v)
    Cp[(r0 + v + hi) * NN + c0 + l16] = make_float2(crr[v], cii[v]);
}

// ---------------------------------------------------------------------------
// Inverse FFT pass B, real output accumulated into d_out:
// out[y,x'] += Re( sum_x A[y,x] * Wi[x',x] )
// ---------------------------------------------------------------------------
__global__ __launch_bounds__(32) void k_inv_real_acc(
    const float2* __restrict__ A, const float* __restrict__ Wre,
    const float* __restrict__ Wim, float* __restrict__ out) {
  const int img = blockIdx.y;
  const int r0 = (blockIdx.x >> 3) * 16, c0 = (blockIdx.x & 7) * 16;
  const int l = threadIdx.x;
  const int l16 = l & 15;
  const int koff = (l >> 4) * 2;
  const float2* Ap = A + (size_t)img * NP;
  v8f acc = {};
  for (int t0 = 0; t0 < NN; t0 += 4) {
    const int t = t0 + koff;
    float2 a0 = Ap[(r0 + l16) * NN + t];
    float2 a1 = Ap[(r0 + l16) * NN + t + 1];
    v2f are, naim;
    are.x = a0.x;   are.y = a1.x;
    naim.x = -a0.y; naim.y = -a1.y;
    v2f bre, bim;
    bre.x = Wre[(c0 + l16) * NN + t];
    bre.y = Wre[(c0 + l16) * NN + t + 1];
    bim.x = Wim[(c0 + l16) * NN + t];
    bim.y = Wim[(c0 + l16) * NN + t + 1];
    acc = wmma4(are, bre, acc);
    acc = wmma4(naim, bim, acc);
  }
  float* op = out + (size_t)img * NP;
  const int hi = (l >> 4) * 8;
  for (int v = 0; v < 8; ++v)
    op[(r0 + v + hi) * NN + c0 + l16] += acc[v];
}

// ---------------------------------------------------------------------------
// x2 as WMMA GEMM. For plane (b,k):
//   out[y][x'] = Re( sum_{t=(c,p)} E1[c,k,p,y] * G2[b,c,k,p,x'] )
// E1 already carries the 1/N^2 scale. t and t+1 share c (koff even).
// Writes (initializes) d_out; x1 is accumulated on top afterwards.
// ---------------------------------------------------------------------------
__global__ __launch_bounds__(32) void k_x2_gemm(
    const float2* __restrict__ G2, const float2* __restrict__ E1,
    float* __restrict__ out) {
  const int plane = blockIdx.y;  // b*32+k
  const int b = plane >> 5, k = plane & 31;
  const int r0 = (blockIdx.x >> 3) * 16, c0 = (blockIdx.x & 7) * 16;
  const int l = threadIdx.x;
  const int l16 = l & 15;
  const int koff = (l >> 4) * 2;
  v8f acc = {};
  for (int t0 = 0; t0 < NN; t0 += 4) {
    const int t = t0 + koff;  // t = c*4 + p
    const int c = t >> 2, p = t & 3;
    const size_t ea = (size_t)((c * NC + k) * NM1 + p) * NN + (r0 + l16);
    float2 e0 = E1[ea];
    float2 e1 = E1[ea + NN];  // p+1
    const size_t gb = (size_t)(b * 4096 + c * NN + k * 4 + p) * NN + c0 + l16;
    float2 g0 = G2[gb];
    float2 g1 = G2[gb + NN];  // p+1
    v2f are, naim, bre, bim;
    are.x = e0.x;   are.y = e1.x;
    naim.x = -e0.y; naim.y = -e1.y;
    bre.x = g0.x;   bre.y = g1.x;
    bim.x = g0.y;   bim.y = g1.y;
    acc = wmma4(are, bre, acc);
    acc = wmma4(naim, bim, acc);
  }
  float* op = out + (size_t)plane * NP;
  const int hi = (l >> 4) * 8;
  for (int v = 0; v < 8; ++v)
    op[(r0 + v + hi) * NN + c0 + l16] = acc[v];
}

// ---------------------------------------------------------------------------
// res1[b,k,o,x] = sum_i alpha[b,i,o,x] * Hsum[i,k,o,x]; one block per (o,x).
// Hsum[i,k] = sum_p A1[i,k,p,o] * W2[i,k,p,x] built into LDS (W2 pre-folds
// the q contraction, so only 4 cMACs per entry here).
// ---------------------------------------------------------------------------
__global__ __launch_bounds__(256) void k_res1(
    const float2* __restrict__ A1, const float2* __restrict__ W2,
    const float2* __restrict__ alpha, float2* __restrict__ res1) {
  __shared__ float2 Hs[NC * NC];  // 8 KB
  __shared__ float2 As[NB * NC];  // 2 KB
  const int fid = blockIdx.x;
  const int o = fid >> 7, x = fid & 127;
  const int tid = threadIdx.x;
  for (int j = 0; j < 4; ++j) {
    const int ik = tid + 256 * j;  // ik == i*32+k
    float2 acc = make_float2(0.f, 0.f);
    for (int p = 0; p < NM1; ++p)
      acc = cfma(A1[(ik * NM1 + p) * NN + o], W2[(ik * NM1 + p) * NN + x], acc);
    Hs[ik] = acc;
  }
  As[tid] = alpha[(size_t)tid * NP + o * NN + x];  // tid == b*32+i plane
  __syncthreads();
  const int b = tid >> 5, k = tid & 31;
  float2 r = make_float2(0.f, 0.f);
  for (int i = 0; i < NC; ++i) r = cfma(As[b * NC + i], Hs[i * NC + k], r);
  res1[(size_t)(b * NC + k) * NP + o * NN + x] = r;
}

// U[b,i,k,p,q] = sum_x T1[b,i,k,p,x] * A2[i,k,q,x]
__global__ __launch_bounds__(256) void k_U(const float2* __restrict__ T1,
                                           const float2* __restrict__ A2,
                                           float2* __restrict__ U) {
  const int idx = blockIdx.x * 256 + threadIdx.x;
  if (idx >= NB * NC * NC * NM1 * NM2) return;  // 163840
  const int q = idx % 5;
  int r = idx / 5;
  const int p = r & 3;  r >>= 2;
  const int k = r & 31; r >>= 5;
  const int i = r & 31;
  const int b = r >> 5;
  const float2* tp = T1 + (size_t)(((b * NC + i) * NC + k) * NM1 + p) * NN;
  const float2* a2p = A2 + ((i * NC + k) * NM2 + q) * NN;
  float2 acc = make_float2(0.f, 0.f);
  for (int x = 0; x < NN; ++x) acc = cfma(tp[x], a2p[x], acc);
  U[idx] = acc;
}

// res2[b,k,p,q] = sum_i wr[i,k,p,q] * U[b,i,k,p,q]
__global__ __launch_bounds__(256) void k_res2(const float2* __restrict__ U,
                                              const float* __restrict__ wr_re,
                                              const float* __restrict__ wr_im,
                                              float2* __restrict__ res2) {
  const int idx = blockIdx.x * 256 + threadIdx.x;
  if (idx >= NB * NC * NM1 * NM2) return;  // 5120
  const int q = idx % 5;
  int r = idx / 5;
  const int p = r & 3;  r >>= 2;
  const int k = r & 31;
  const int b = r >> 5;
  float2 acc = make_float2(0.f, 0.f);
  for (int i = 0; i < NC; ++i) {
    const int wri = (i * NC + k) * 20 + p * 5 + q;
    float2 w = make_float2(wr_re[wri], wr_im[wri]);
    acc = cfma(w, U[(((b * NC + i) * NC + k) * NM1 + p) * NM2 + q], acc);
  }
  res2[idx] = acc;  // layout ((b*32+k)*4+p)*5+q
}

// G2[b,c,k,p,x'] = sum_q res2[b,c,p,q] * E2[c,k,q,x']
__global__ __launch_bounds__(256) void k_G2(const float2* __restrict__ res2,
                                            const float2* __restrict__ E2,
                                            float2* __restrict__ G2) {
  const int idx = blockIdx.x * 256 + threadIdx.x;  // 4,194,304
  const int x = idx & 127;
  const int p = (idx >> 7) & 3;
  const int k = (idx >> 9) & 31;
  const int c = (idx >> 14) & 31;
  const int b = idx >> 19;
  float2 acc = make_float2(0.f, 0.f);
  for (int q = 0; q < NM2; ++q)
    acc = cfma(res2[((b * NC + c) * NM1 + p) * NM2 + q],
               E2[((c * NC + k) * NM2 + q) * NN + x], acc);
  G2[idx] = acc;
}

// ---------------------------------------------------------------------------
extern "C" void kernel_launch(void* const* d_in, const int* in_sizes, int n_in,
                              void* d_out, int out_size, void* d_ws,
                              size_t ws_size, hipStream_t stream) {
  (void)in_sizes; (void)n_in; (void)out_size; (void)ws_size;
  const float* x      = (const float*)d_in[0];
  const float* wp1_re = (const float*)d_in[1];
  const float* wp1_im = (const float*)d_in[2];
  const float* wp2_re = (const float*)d_in[3];
  const float* wp2_im = (const float*)d_in[4];
  const float* wr_re  = (const float*)d_in[5];
  const float* wr_im  = (const float*)d_in[6];
  const float* ty     = (const float*)d_in[7];
  const float* tx     = (const float*)d_in[8];
  float* out = (float*)d_out;

  // Workspace layout (floats); total ~39.7M floats = ~159 MB (fits L2).
  float* ws = (float*)d_ws;
  size_t off = 0;
  float* Wf_re = ws + off; off += NP;
  float* Wf_im = ws + off; off += NP;
  float* Wi_re = ws + off; off += NP;
  float* Wi_im = ws + off; off += NP;
  float2* A1   = (float2*)(ws + off); off += 2 * (size_t)NC * NC * NM1 * NN;
  float2* A2   = (float2*)(ws + off); off += 2 * (size_t)NC * NC * NM2 * NN;
  float2* E1   = (float2*)(ws + off); off += 2 * (size_t)NC * NC * NM1 * NN;
  float2* E2   = (float2*)(ws + off); off += 2 * (size_t)NC * NC * NM2 * NN;
  float2* W2   = (float2*)(ws + off); off += 2 * (size_t)NC * NC * NM1 * NN;
  float2* U    = (float2*)(ws + off); off += 2 * (size_t)NB * NC * NC * NM1 * NM2;
  float2* res2 = (float2*)(ws + off); off += 2 * (size_t)NB * NC * NM1 * NM2;
  float2* big0 = (float2*)(ws + off); off += 2 * (size_t)NIMG * NP;  // fwd/inv tmp
  float2* big1 = (float2*)(ws + off); off += 2 * (size_t)NIMG * NP;  // alpha
  float2* big2 = (float2*)(ws + off); off += 2 * (size_t)NIMG * NP;  // res1
  float2* big3 = (float2*)(ws + off); off += 2 * (size_t)NIMG * NP;  // T1 / G2

  // 0) Twiddle + pole tables; fold q-contraction into W2.
  k_tables<<<dim3((NC * NC * NM2 * NN + 255) / 256), 256, 0, stream>>>(
      wp1_re, wp1_im, wp2_re, wp2_im, ty, tx, Wf_re, Wf_im, Wi_re, Wi_im, A1,
      A2, E1, E2);
  k_W2<<<dim3((NC * NC * NM1 * NN) / 256), 256, 0, stream>>>(A2, wr_re, wr_im,
                                                             W2);

  const dim3 gmm(64, NIMG);  // 8x8 tiles of 16x16 per 128x128 plane, 1 wave ea.

  // 1) alpha = fft2(x): rows (real*complex) then cols (complex*complex).
  k_fwd_rows<<<gmm, 32, 0, stream>>>(x, Wf_re, Wf_im, big0);
  k_gemm_cc<<<gmm, 32, 0, stream>>>(Wf_re, Wf_im, big0, big1);

  // 2) res1 via on-the-fly Hsum (A1 x W2 in LDS).
  k_res1<<<dim3(NP), 256, 0, stream>>>(A1, W2, big1, big2);

  // 3) res2 factorized: T1 (WMMA GEMM) -> U -> res2.
  k_T1_gemm<<<gmm, 32, 0, stream>>>(A1, big1, big3);
  k_U<<<dim3((NB * NC * NC * NM1 * NM2 + 255) / 256), 256, 0, stream>>>(big3,
                                                                        A2, U);
  k_res2<<<dim3((NB * NC * NM1 * NM2 + 255) / 256), 256, 0, stream>>>(
      U, wr_re, wr_im, res2);

  // 4) x2 -> d_out (initializes output). big3 (T1) is dead, reuse for G2.
  k_G2<<<dim3((NB * NC * NC * NM1 * NN) / 256), 256, 0, stream>>>(res2, E2,
                                                                  big3);
  k_x2_gemm<<<gmm, 32, 0, stream>>>(big3, E1, out);

  // 5) x1 = Re(ifft2(res1)) accumulated into d_out.
  k_gemm_cc<<<gmm, 32, 0, stream>>>(Wi_re, Wi_im, big2, big0);
  k_inv_real_acc<<<gmm, 32, 0, stream>>>(big0, Wi_re, Wi_im, out);
}